// PositionEncoder_63866163691554
// MI455X (gfx1250) — compile-verified
//
#include <hip/hip_runtime.h>
#include <stdint.h>

// out[b,s,d] = xs[b,s,d] + P[s,d]
//   P[s,2i]   = sin(s * D * 10000^(-2i))
//   P[s,2i+1] = cos(s * D * 10000^(-2i))
// Shapes fixed by the reference: B=8, S=4096, D=1024, f32.

#define B_LEN 8
#define S_LEN 4096
#define D_LEN 1024
#define ROWS_PER_BLK 4
#define THREADS 256
#define ROW_BYTES (D_LEN * 4u)                       // 4096 B
#define TILE_BYTES (ROWS_PER_BLK * ROW_BYTES)        // 16 KiB per buffer
#define BATCH_BYTES (S_LEN * D_LEN * 4u)             // 16 MiB

typedef float v4f __attribute__((ext_vector_type(4)));

// Issue 4 async B128 loads (one tile row per load per lane) global -> LDS.
// GVS addressing: saddr = xs base (SGPR pair), vaddr = 32-bit byte offset.
__device__ __forceinline__ void issue_tile(const float* xs, unsigned gbase, unsigned lbase) {
#pragma unroll
  for (int r = 0; r < ROWS_PER_BLK; ++r) {
    const unsigned voff = gbase + (unsigned)r * ROW_BYTES;
    const unsigned loff = lbase + (unsigned)r * ROW_BYTES;
    asm volatile("global_load_async_to_lds_b128 %0, %1, %2 th:TH_LOAD_NT"
                 :
                 : "v"(loff), "v"(voff), "s"(xs)
                 : "memory");
  }
}

__global__ __launch_bounds__(THREADS) void pe_add_kernel(const float* __restrict__ xs,
                                                         float* __restrict__ out) {
  __shared__ alignas(16) float buf[2][ROWS_PER_BLK * D_LEN];  // 32 KiB, double buffered

  const unsigned tid = threadIdx.x;                // 0..255, owns d = 4*tid .. 4*tid+3
  const unsigned s0  = blockIdx.x * ROWS_PER_BLK;  // first row of this block's tile

  // ---- Positional values for this thread's 4 columns x 4 rows, computed ONCE,
  //      reused for all 8 batches (8x fewer transcendentals).
  const float i0 = (float)(2u * tid);
  const float i1 = i0 + 1.0f;
  const float kneg = -2.0f * 13.287712379549449f;  // -2*log2(10000)
  const float sc0 = (float)D_LEN * exp2f(kneg * i0);  // underflows to 0 for large i (matches ref)
  const float sc1 = (float)D_LEN * exp2f(kneg * i1);

  v4f P[ROWS_PER_BLK];
#pragma unroll
  for (int r = 0; r < ROWS_PER_BLK; ++r) {
    const float pos = (float)(s0 + (unsigned)r);
    const float a0 = pos * sc0;
    const float a1 = pos * sc1;
    v4f p;
    p.x = __sinf(a0);
    p.y = __cosf(a0);
    p.z = __sinf(a1);
    p.w = __cosf(a1);
    P[r] = p;
  }

  // Flat-pointer low 32 bits of an LDS object == wave-relative LDS byte address
  // (aperture lives in the high 32 bits), which is what the async-load VDST wants.
  const unsigned lds_base = (unsigned)(uintptr_t)&buf[0][0];
  const unsigned laneByte = tid * 16u;                 // this lane's 16B within a row
  const unsigned rowBase  = s0 * ROW_BYTES;            // byte offset of tile inside one batch

  // Prologue: stage batch 0 into buffer 0.
  issue_tile(xs, 0u * BATCH_BYTES + rowBase + laneByte, lds_base + 0u * TILE_BYTES + laneByte);

#pragma unroll
  for (unsigned b = 0; b < B_LEN; ++b) {
    const unsigned cur = b & 1u;

    if (b + 1u < B_LEN) {
      // Make sure our ds_load reads of buf[cur^1] (previous iteration) fully
      // drained before the TA starts overwriting that buffer (WAR vs async writes).
      asm volatile("s_wait_dscnt 0" ::: "memory");
      issue_tile(xs, (b + 1u) * BATCH_BYTES + rowBase + laneByte,
                 lds_base + (cur ^ 1u) * TILE_BYTES + laneByte);
      // Async loads complete in order per wave: <=4 outstanding means the 4
      // loads for batch b have landed; batch b+1's 4 may still be in flight.
      asm volatile("s_wait_asynccnt 4" ::: "memory");
    } else {
      asm volatile("s_wait_asynccnt 0" ::: "memory");
    }

    // Each lane consumes exactly the 16B it requested -> no barriers needed.
#pragma unroll
    for (int r = 0; r < ROWS_PER_BLK; ++r) {
      const v4f x = *(const v4f*)&buf[cur][(unsigned)r * D_LEN + tid * 4u];
      v4f v;
      v.x = x.x + P[r].x;
      v.y = x.y + P[r].y;
      v.z = x.z + P[r].z;
      v.w = x.w + P[r].w;
      const unsigned voff = b * BATCH_BYTES + rowBase + (unsigned)r * ROW_BYTES + laneByte;
      __builtin_nontemporal_store(v, (v4f*)((char*)out + (size_t)voff));
    }
  }
}

extern "C" void kernel_launch(void* const* d_in, const int* in_sizes, int n_in,
                              void* d_out, int out_size, void* d_ws, size_t ws_size,
                              hipStream_t stream) {
  (void)in_sizes; (void)n_in; (void)ws_size; (void)d_ws; (void)out_size;
  const float* xs = (const float*)d_in[0];
  float* out = (float*)d_out;
  dim3 grid(S_LEN / ROWS_PER_BLK);  // 1024 blocks x 256 threads, covers S x D exactly
  pe_add_kernel<<<grid, THREADS, 0, stream>>>(xs, out);
}